// ChebyshevConv_5669356834169
// MI455X (gfx1250) — compile-verified
//
#include <hip/hip_runtime.h>
#include <hip/hip_bf16.h>

typedef __attribute__((ext_vector_type(2))) float v2f;
typedef __attribute__((ext_vector_type(8))) float v8f;

#define EPSF 1e-10f

// ---------------- degree accumulation ----------------
__global__ void cheb_deg_kernel(const int* __restrict__ ei, const float* __restrict__ ew,
                                float* __restrict__ deg_in, float* __restrict__ deg_out, int E) {
    int e = blockIdx.x * blockDim.x + threadIdx.x;
    if (e >= E) return;
    int s = ei[2 * e];
    int d = ei[2 * e + 1];
    float w = ew[e];
    atomicAdd(deg_in + s, w);
    atomicAdd(deg_out + d, w);
}

// ---------------- 1/(sqrt(deg)+eps) ----------------
__global__ void cheb_invsqrt_kernel(const float* __restrict__ deg_in, const float* __restrict__ deg_out,
                                    float* __restrict__ inv_in, float* __restrict__ inv_out, int N) {
    int i = blockIdx.x * blockDim.x + threadIdx.x;
    if (i >= N) return;
    inv_in[i]  = 1.0f / (sqrtf(deg_in[i])  + EPSF);
    inv_out[i] = 1.0f / (sqrtf(deg_out[i]) + EPSF);
}

// ---------------- copy x into feats columns [0,64) ----------------
__global__ void cheb_copyx_kernel(const float* __restrict__ x, float* __restrict__ feats, int N) {
    int t = blockIdx.x * blockDim.x + threadIdx.x;   // N*16 threads, float4 each
    if (t >= N * 16) return;
    int n = t >> 4;
    int g = (t & 15) * 4;
    float4 v = *(const float4*)(x + (size_t)n * 64 + g);
    *(float4*)(feats + (size_t)n * 256 + g) = v;
}

// ---------------- edge scatter: acc[dst] += h[src]*inv_in[src]*w ----------------
__global__ void cheb_scatter_kernel(const int* __restrict__ ei, const float* __restrict__ ew,
                                    const float* __restrict__ inv_in, const float* __restrict__ feats,
                                    int col, float* __restrict__ acc, int E) {
    int t = blockIdx.x * blockDim.x + threadIdx.x;   // E*16 threads, 4 feats each
    if (t >= E * 16) return;
    int e = t >> 4;
    int g = (t & 15) * 4;
    int s = ei[2 * e];
    int d = ei[2 * e + 1];
    float scale = ew[e] * inv_in[s];
    float4 h = *(const float4*)(feats + (size_t)s * 256 + col + g);
    float* dst = acc + (size_t)d * 64 + g;
    atomicAdd(dst + 0, h.x * scale);
    atomicAdd(dst + 1, h.y * scale);
    atomicAdd(dst + 2, h.z * scale);
    atomicAdd(dst + 3, h.w * scale);
}

// ------- node update: h_new = h_prev - acc*inv_out ; optionally 2*h_new - h_prev2 -------
__global__ void cheb_update_kernel(float* __restrict__ feats, const float* __restrict__ acc,
                                   const float* __restrict__ inv_out,
                                   int col_prev, int col_out, int col_pm2, int doubled, int N) {
    int t = blockIdx.x * blockDim.x + threadIdx.x;   // N*16 threads, float4 each
    if (t >= N * 16) return;
    int n = t >> 4;
    int g = (t & 15) * 4;
    float io = inv_out[n];
    size_t row = (size_t)n * 256;
    float4 h = *(const float4*)(feats + row + col_prev + g);
    float4 a = *(const float4*)(acc + (size_t)n * 64 + g);
    float4 v;
    v.x = h.x - a.x * io;
    v.y = h.y - a.y * io;
    v.z = h.z - a.z * io;
    v.w = h.w - a.w * io;
    if (doubled) {
        float4 p = *(const float4*)(feats + row + col_pm2 + g);
        v.x = 2.0f * v.x - p.x;
        v.y = 2.0f * v.y - p.y;
        v.z = 2.0f * v.z - p.z;
        v.w = 2.0f * v.w - p.w;
    }
    *(float4*)(feats + row + col_out + g) = v;
}

// ---------------- WMMA fp32 GEMM + bias + relu ----------------
// out[N,64] = relu(feats[N,256] @ W[256,64] + b)
// One wave per 16-row tile; 4 accumulators cover all 64 output columns.
// A fragment (16x4 f32): lane = (k>=2 ? 16 : 0) + m, vgpr = k&1
// B fragment (4x16 f32): lane = (k>=2 ? 16 : 0) + n, vgpr = k&1
// C/D (16x16 f32): lanes 0-15 -> M=v, N=lane; lanes 16-31 -> M=v+8, N=lane-16
__global__ __launch_bounds__(256) void cheb_gemm_kernel(
    const float* __restrict__ feats, const float* __restrict__ W,
    const float* __restrict__ bias, float* __restrict__ out, int N) {
    const int lane = threadIdx.x & 31;
    const int wave = threadIdx.x >> 5;
    const int m0 = (blockIdx.x * 8 + wave) * 16;
    if (m0 >= N) return;                       // wave-uniform: EXEC stays all-ones

    const int mrow = lane & 15;
    const int koff = (lane >> 4) * 2;          // 0 or 2

    v8f acc[4] = {};

    const float* arow = feats + (size_t)(m0 + mrow) * 256 + koff;
    for (int k = 0; k < 256; k += 4) {
        v2f a;
        a.x = arow[k];
        a.y = arow[k + 1];
#pragma unroll
        for (int t = 0; t < 4; ++t) {
            int n = t * 16 + mrow;
            v2f bb;
            bb.x = W[(k + koff) * 64 + n];
            bb.y = W[(k + koff + 1) * 64 + n];
            acc[t] = __builtin_amdgcn_wmma_f32_16x16x4_f32(
                /*neg_a=*/false, a, /*neg_b=*/false, bb,
                /*c_mod=*/(short)0, acc[t], /*reuse_a=*/false, /*reuse_b=*/false);
        }
    }

    const int rbase = m0 + ((lane >> 4) * 8);
#pragma unroll
    for (int t = 0; t < 4; ++t) {
        int col = t * 16 + mrow;
        float bv = bias[col];
#pragma unroll
        for (int v = 0; v < 8; ++v) {
            float val = acc[t][v] + bv;
            out[(size_t)(rbase + v) * 64 + col] = val > 0.0f ? val : 0.0f;
        }
    }
}

extern "C" void kernel_launch(void* const* d_in, const int* in_sizes, int n_in,
                              void* d_out, int out_size, void* d_ws, size_t ws_size,
                              hipStream_t stream) {
    const float* x  = (const float*)d_in[0];
    const int*   ei = (const int*)d_in[1];
    const float* ew = (const float*)d_in[2];
    const float* W  = (const float*)d_in[3];
    const float* b  = (const float*)d_in[4];
    float* out = (float*)d_out;

    const int N = in_sizes[0] / 64;
    const int E = in_sizes[2];
    const int Np = ((N + 63) / 64) * 64;

    float* ws = (float*)d_ws;
    float* deg_in  = ws;
    float* deg_out = ws + (size_t)Np;
    float* inv_in  = ws + (size_t)2 * Np;
    float* inv_out = ws + (size_t)3 * Np;
    float* acc     = ws + (size_t)4 * Np;               // Np*64 floats
    float* feats   = ws + (size_t)4 * Np + (size_t)Np * 64;  // N*256 floats

    const int TB = 256;

    // degrees
    hipMemsetAsync(deg_in, 0, (size_t)2 * Np * sizeof(float), stream);
    cheb_deg_kernel<<<(E + TB - 1) / TB, TB, 0, stream>>>(ei, ew, deg_in, deg_out, E);
    cheb_invsqrt_kernel<<<(N + TB - 1) / TB, TB, 0, stream>>>(deg_in, deg_out, inv_in, inv_out, N);
    cheb_copyx_kernel<<<(N * 16 + TB - 1) / TB, TB, 0, stream>>>(x, feats, N);

    // K=3 Chebyshev propagation steps
    for (int i = 0; i < 3; ++i) {
        hipMemsetAsync(acc, 0, (size_t)Np * 64 * sizeof(float), stream);
        cheb_scatter_kernel<<<(E * 16 + TB - 1) / TB, TB, 0, stream>>>(
            ei, ew, inv_in, feats, i * 64, acc, E);
        cheb_update_kernel<<<(N * 16 + TB - 1) / TB, TB, 0, stream>>>(
            feats, acc, inv_out,
            /*col_prev=*/i * 64, /*col_out=*/(i + 1) * 64,
            /*col_pm2=*/(i - 1) * 64, /*doubled=*/(i > 0) ? 1 : 0, N);
    }

    // fused GEMM + bias + relu with fp32 WMMA
    cheb_gemm_kernel<<<(N + 127) / 128, TB, 0, stream>>>(feats, W, b, out, N);
}